// ScaledDotProductAttention_78288663872424
// MI455X (gfx1250) — compile-verified
//
#include <hip/hip_runtime.h>

// ---------------------------------------------------------------------------
// Fused scaled-dot-product attention for MI455X (gfx1250), fp32 end-to-end.
// scores = (Q K^T) * 1/8, mask-fill -1e9, softmax, P written to attn_prob,
// context = P V.  One block = one (b*h, 16-row query tile).
// Matrix math: V_WMMA_F32_16X16X4_F32 (exact fp32, chip is HBM-bound here).
// ---------------------------------------------------------------------------

typedef float v2f __attribute__((ext_vector_type(2)));
typedef float v8f __attribute__((ext_vector_type(8)));

#define S_LEN 2048
#define D_HID 64
#define BH 64                    // B*H = 4*16
#define QT 16                    // query rows per block
#define NKT (S_LEN / 16)         // 128 key tiles
#define WAVES 8
#define SROW (S_LEN + 4)         // padded LDS row stride (floats)
#define SCALE 0.125f
#define MASK_FILL -1e9f

__global__ __launch_bounds__(256, 1)
void sdpa_fused_f32_wmma(const float* __restrict__ Q,
                         const float* __restrict__ K,
                         const float* __restrict__ V,
                         const unsigned char* __restrict__ mask,
                         float* __restrict__ ctx,
                         float* __restrict__ prob) {
  extern __shared__ float smem[];
  float* sS     = smem;                 // [16][SROW]  score / prob strip
  float* red    = sS + 16 * SROW;       // [16][16]    reduction scratch
  float* rowMax = red + 256;            // [16]
  float* rowSum = rowMax + 16;          // [16]
  float* cbuf   = rowSum + 16;          // [8][8*32]   per-wave partial C

  const int tid  = threadIdx.x;
  const int wave = tid >> 5;
  const int lane = tid & 31;
  const int half = lane >> 4;           // 0: lanes 0-15, 1: lanes 16-31
  const int l16  = lane & 15;

  const int qt = blockIdx.x;            // 0..127 query tile
  const int bh = blockIdx.y;            // 0..63  fused batch*head
  const int q0 = qt * QT;
  const size_t qkBase   = (size_t)bh * S_LEN * D_HID;
  const size_t probBase = (size_t)bh * S_LEN * S_LEN + (size_t)q0 * S_LEN;

  // ---- Phase A: scores = scale * Q K^T, masked, into LDS -------------------
  // A-operand (16x4 f32): lane m=l16, this lane holds K = 4c + 2*half + {0,1}
  const float* qrow = Q + qkBase + (size_t)(q0 + l16) * D_HID + half * 2;
  v2f aQ[16];
#pragma unroll
  for (int c = 0; c < 16; ++c) {
    aQ[c].x = qrow[4 * c + 0];
    aQ[c].y = qrow[4 * c + 1];
  }

  for (int kt = wave; kt < NKT; kt += WAVES) {
    const float* krow = K + qkBase + (size_t)(kt * 16 + l16) * D_HID;
    if (kt + WAVES < NKT)  // emits global_prefetch_b8 for the next K tile
      __builtin_prefetch(K + qkBase + (size_t)((kt + WAVES) * 16 + l16) * D_HID, 0, 0);

    v8f acc = {0.f, 0.f, 0.f, 0.f, 0.f, 0.f, 0.f, 0.f};
#pragma unroll
    for (int c = 0; c < 16; ++c) {
      // B-operand (4x16): column n=l16; VGPR0 = B[half][n], VGPR1 = B[2+half][n]
      const float4 kv = *(const float4*)(krow + 4 * c);
      v2f bK;
      bK.x = half ? kv.y : kv.x;
      bK.y = half ? kv.w : kv.z;
      acc = __builtin_amdgcn_wmma_f32_16x16x4_f32(
          false, aQ[c], false, bK, (short)0, acc, false, false);
    }

    // C/D layout: VGPR r holds (M = r + 8*half, N = l16)
#pragma unroll
    for (int r = 0; r < 8; ++r) {
      const int m = r + half * 8;
      const int n = kt * 16 + l16;
      const unsigned char mb = mask[probBase + (size_t)m * S_LEN + n];
      sS[m * SROW + n] = mb ? MASK_FILL : acc[r] * SCALE;
    }
  }
  __syncthreads();

  // ---- Phase B: row softmax over LDS strip, write attn_prob ----------------
  const int row = tid >> 4;       // 16 threads per row
  const int c16 = tid & 15;

  float mx = -3.4e38f;
  for (int j = c16; j < S_LEN; j += 16) mx = fmaxf(mx, sS[row * SROW + j]);
  red[tid] = mx;
  __syncthreads();
  if (c16 == 0) {
    float m2 = red[row * 16];
#pragma unroll
    for (int i = 1; i < 16; ++i) m2 = fmaxf(m2, red[row * 16 + i]);
    rowMax[row] = m2;
  }
  __syncthreads();

  const float rmx = rowMax[row];
  float sum = 0.f;
  for (int j = c16; j < S_LEN; j += 16) {
    const float e = __expf(sS[row * SROW + j] - rmx);
    sS[row * SROW + j] = e;
    sum += e;
  }
  red[tid] = sum;
  __syncthreads();
  if (c16 == 0) {
    float s2 = 0.f;
#pragma unroll
    for (int i = 0; i < 16; ++i) s2 += red[row * 16 + i];
    rowSum[row] = s2;
  }
  __syncthreads();

  const float rinv = 1.0f / rowSum[row];
  float* probRow = prob + probBase + (size_t)row * S_LEN;
  for (int j = c16; j < S_LEN; j += 16) {
    const float p = sS[row * SROW + j] * rinv;
    sS[row * SROW + j] = p;   // keep normalized P in LDS for phase C
    probRow[j] = p;           // mandatory 1.07 GB output stream
  }
  __syncthreads();

  // ---- Phase C: context = P(16x2048) . V(2048x64) --------------------------
  // wave -> (N tile = wave&3, K half = wave>>2); reduce pairs through LDS.
  const int vt = wave & 3;
  const int k0 = (wave >> 2) * (S_LEN / 2);
  v8f accC = {0.f, 0.f, 0.f, 0.f, 0.f, 0.f, 0.f, 0.f};

  const float* vcol = V + qkBase + vt * 16 + l16;
#pragma unroll 4
  for (int kk = 0; kk < S_LEN / 2; kk += 4) {
    const int kidx = k0 + kk;
    // A-operand: P chunk 16x4 straight from LDS (contiguous float2 per lane)
    const v2f aP = *(const v2f*)(&sS[l16 * SROW + kidx + half * 2]);
    // B-operand: V chunk 4x16
    const float* vp = vcol + (size_t)(kidx + half) * D_HID;
    v2f bV;
    bV.x = vp[0];
    bV.y = vp[2 * D_HID];
    accC = __builtin_amdgcn_wmma_f32_16x16x4_f32(
        false, aP, false, bV, (short)0, accC, false, false);
  }

#pragma unroll
  for (int r = 0; r < 8; ++r) cbuf[wave * 256 + r * 32 + lane] = accC[r];
  __syncthreads();

  if (wave < 4) {
#pragma unroll
    for (int r = 0; r < 8; ++r) {
      const float v0 = cbuf[wave * 256 + r * 32 + lane] +
                       cbuf[(wave + 4) * 256 + r * 32 + lane];
      const int m = r + half * 8;
      ctx[qkBase + (size_t)(q0 + m) * D_HID + vt * 16 + l16] = v0;
    }
  }
}

extern "C" void kernel_launch(void* const* d_in, const int* in_sizes, int n_in,
                              void* d_out, int out_size, void* d_ws, size_t ws_size,
                              hipStream_t stream) {
  (void)in_sizes; (void)n_in; (void)d_ws; (void)ws_size; (void)out_size;

  const float* Q = (const float*)d_in[0];
  const float* K = (const float*)d_in[1];
  const float* V = (const float*)d_in[2];
  const unsigned char* mask = (const unsigned char*)d_in[3];  // jnp.bool_ = 1 byte

  float* ctx  = (float*)d_out;                                 // [4,16,2048,64]
  float* prob = ctx + (size_t)BH * S_LEN * D_HID;              // [4,16,2048,2048]

  const size_t lds_bytes =
      (size_t)(16 * SROW + 256 + 32 + 8 * 256) * sizeof(float); // ~140.7 KB

  dim3 grid(S_LEN / QT, BH);   // 128 x 64 blocks
  dim3 block(256);             // 8 waves (wave32)
  sdpa_fused_f32_wmma<<<grid, block, lds_bytes, stream>>>(Q, K, V, mask, ctx, prob);
}